// LRPEMultiHeadAttention_81578608820229
// MI455X (gfx1250) — compile-verified
//
#include <hip/hip_runtime.h>
#include <hip/hip_bf16.h>
#include <math.h>

// Problem constants (fixed by the reference).
#define BB 8
#define NN 1024
#define MM 1024
#define DD 256
#define HH 4
#define CC 64
#define PP 128

typedef __attribute__((ext_vector_type(2))) float v2f;
typedef __attribute__((ext_vector_type(8))) float v8f;

#define V8F_ZERO {0.f,0.f,0.f,0.f,0.f,0.f,0.f,0.f}

// D = A(16x4) * B(4x16) + C, fp32 all the way (matches fp32 reference precision).
static __device__ __forceinline__ v8f wmma_f32(v2f a, v2f b, v8f c) {
    return __builtin_amdgcn_wmma_f32_16x16x4_f32(
        /*neg_a=*/false, a, /*neg_b=*/false, b,
        /*c_mod=*/(short)0, c, /*reuse_a=*/false, /*reuse_b=*/false);
}

// ---------------------------------------------------------------------------
// Kernel 1: out = X @ W^T + bias, written head-major as (B, H, L, C).
// grid = (B*L/16, D/64), block = 128 (4 waves); each wave owns a 16x16 tile.
// A tile (16 rows x 256, contiguous) staged in LDS; W rows read from L2.
// ---------------------------------------------------------------------------
__global__ void qkv_proj_kernel(const float* __restrict__ X,
                                const float* __restrict__ W,
                                const float* __restrict__ bias,
                                float* __restrict__ out) {
    __shared__ float Alds[16 * DD];  // 16 KB

    const int tid  = threadIdx.x;
    const int wave = tid >> 5;
    const int lane = tid & 31;
    const int row0 = blockIdx.x * 16;   // global row tile (over B*L)
    const int col0 = blockIdx.y * 64;   // output-feature base (4 waves x 16)

    {   // cooperative stage: 16x256 fp32 = 1024 float4, 8 per thread
        const float4* src = (const float4*)(X + (size_t)row0 * DD);
        float4* dst = (float4*)Alds;
        for (int i = tid; i < 16 * DD / 4; i += 128) dst[i] = src[i];
    }
    __syncthreads();

    const int mrow = lane & 15;          // M row (lanes 0-15 / 16-31 share)
    const int kk2  = (lane >> 4) * 2;    // K sub-pair selected by lane group
    const int mg   = (lane >> 4) * 8;    // C/D row group

    const float* ap = Alds + mrow * DD + kk2;
    // out[m][d] = sum_k X[m][k] * W[d][k]  ->  B[k][n] = W[col0+wave*16+n][k]
    const float* wp = W + (size_t)(col0 + wave * 16 + mrow) * DD + kk2;

    v8f acc = V8F_ZERO;
    for (int k0 = 0; k0 < DD; k0 += 4) {
        v2f a = *(const v2f*)(ap + k0);
        v2f b = *(const v2f*)(wp + k0);
        acc = wmma_f32(a, b, acc);
    }

    const int dcol = col0 + wave * 16 + mrow;
    const int h = dcol >> 6;             // CC == 64
    const int c = dcol & (CC - 1);
    const float bval = bias[dcol];
    for (int v = 0; v < 8; ++v) {
        const int r  = row0 + mg + v;    // global row over B*L
        const int bb = r >> 10;          // L == 1024
        const int rl = r & 1023;
        out[(((size_t)bb * HH + h) * NN + rl) * CC + c] = acc[v] + bval;
    }
}

// ---------------------------------------------------------------------------
// Kernel 2: fully fused LRPE attention for one (b, h, 16-query tile).
// 320KB LDS/WGP lets the entire 16x1024 score tile live in LDS: QK^T, gathered
// positional bias, factors, mask, softmax, attn streaming store and attn@V all
// happen without any HBM intermediate.
// grid = (N/16, H, B), block = 128 (4 waves).
// ---------------------------------------------------------------------------
__global__ void lrpe_attn_kernel(const float* __restrict__ q_ws,
                                 const float* __restrict__ k_ws,
                                 const float* __restrict__ v_ws,
                                 const float* __restrict__ emb_table,
                                 const float* __restrict__ att_factors,
                                 const int*   __restrict__ emb_idx,
                                 const unsigned char* __restrict__ key_masks,
                                 float* __restrict__ hidden_out,
                                 float* __restrict__ attn_out) {
    __shared__ float qlds[16 * CC];   //  4 KB: query tile
    __shared__ float plds[16 * PP];   //  8 KB: positional scores q . emb^T
    __shared__ float slds[16 * MM];   // 64 KB: full score rows for 16 queries

    const int tid  = threadIdx.x;
    const int wave = tid >> 5;
    const int lane = tid & 31;
    const int n0 = blockIdx.x * 16;
    const int h  = blockIdx.y;
    const int b  = blockIdx.z;

    const size_t head   = (size_t)b * HH + h;
    const size_t qbase  = (head * NN + n0) * CC;
    const size_t kvbase = head * MM * CC;

    {   // Phase 0: stage q tile (16x64)
        const float4* src = (const float4*)(q_ws + qbase);
        float4* dst = (float4*)qlds;
        for (int i = tid; i < 16 * CC / 4; i += 128) dst[i] = src[i];
    }
    __syncthreads();

    const int mrow = lane & 15;
    const int kk2  = (lane >> 4) * 2;
    const int mg   = (lane >> 4) * 8;

    // Hoist all 16 A-fragments of the query tile (K=64 -> 16 WMMA K-steps).
    v2f afrag[16];
    for (int i = 0; i < 16; ++i)
        afrag[i] = *(const v2f*)&qlds[mrow * CC + i * 4 + kk2];

    // Phase 1: P = q_tile . emb_bank[h]^T  (16x128), 2 column tiles per wave.
    for (int t = wave; t < PP / 16; t += 4) {
        const int pc = t * 16;
        // B[k][n] = emb_bank[h][pc+n][k] = emb_table[(pc+n)*D + h*C + k]
        const float* eb = emb_table + (size_t)(pc + mrow) * DD + h * CC + kk2;
        v8f acc = V8F_ZERO;
        for (int i = 0; i < 16; ++i) {
            v2f bf = *(const v2f*)(eb + i * 4);
            acc = wmma_f32(afrag[i], bf, acc);
        }
        for (int v = 0; v < 8; ++v)
            plds[(mg + v) * PP + pc + mrow] = acc[v];
    }
    __syncthreads();

    // Phase 2: scores. Each wave owns 16 of the 64 m-chunks (16 wide each).
    for (int mt = wave; mt < MM / 16; mt += 4) {
        const int m0 = mt * 16;
        const float* kp = k_ws + kvbase + (size_t)(m0 + mrow) * CC + kk2;
        v8f acc = V8F_ZERO;
        for (int i = 0; i < 16; ++i) {
            v2f bf = *(const v2f*)(kp + i * 4);
            acc = wmma_f32(afrag[i], bf, acc);
        }
        const int mcol = m0 + mrow;
        const unsigned char km = key_masks[b * MM + mcol];
        for (int v = 0; v < 8; ++v) {
            const int qr = mg + v;
            const size_t off = ((size_t)b * NN + n0 + qr) * MM + mcol;
            // stream-once 64MB: NT loads keep L2 free for K/V/W reuse
            const int   id = __builtin_nontemporal_load(emb_idx + off);
            const float af = __builtin_nontemporal_load(att_factors + off);
            float s = af * ((acc[v] + plds[qr * PP + id]) * 0.125f); // 1/sqrt(64)
            if (km) s = -__builtin_inff();
            slds[qr * MM + mcol] = s;
        }
    }
    __syncthreads();

    // Phase 3: softmax per row (4 rows per wave, wave32 xor-shuffle reduce),
    // then stream the normalized probabilities to HBM with NT stores (128MB).
    for (int qr = wave * 4; qr < wave * 4 + 4; ++qr) {
        float* srow = slds + qr * MM;
        float mx = -__builtin_inff();
        for (int m = lane; m < MM; m += 32) mx = fmaxf(mx, srow[m]);
        for (int o = 16; o > 0; o >>= 1) mx = fmaxf(mx, __shfl_xor(mx, o, 32));
        float sum = 0.f;
        for (int m = lane; m < MM; m += 32) {
            const float e = __expf(srow[m] - mx);
            srow[m] = e;
            sum += e;
        }
        for (int o = 16; o > 0; o >>= 1) sum += __shfl_xor(sum, o, 32);
        const float inv = 1.0f / sum;
        float* aout = attn_out + (head * NN + n0 + qr) * MM;
        for (int m = lane; m < MM; m += 32) {
            const float p = srow[m] * inv;
            srow[m] = p;
            __builtin_nontemporal_store(p, aout + m);
        }
    }
    __syncthreads();

    // Phase 4: hidden = attn_tile(16x1024) @ V(1024x64); wave w owns c-cols
    // [w*16, w*16+16).  A from LDS (ds_load_b64), B from L2-resident V.
    const float* vp = v_ws + kvbase + wave * 16 + mrow;
    v8f acc = V8F_ZERO;
    for (int k0 = 0; k0 < MM; k0 += 4) {
        v2f a = *(const v2f*)&slds[mrow * MM + k0 + kk2];
        v2f bf;
        bf.x = vp[(size_t)(k0 + kk2) * CC];
        bf.y = vp[(size_t)(k0 + kk2 + 1) * CC];
        acc = wmma_f32(a, bf, acc);
    }
    const int c = wave * 16 + mrow;
    for (int v = 0; v < 8; ++v) {
        const int n = n0 + mg + v;
        hidden_out[((size_t)b * NN + n) * DD + h * CC + c] = acc[v];
    }
}

// ---------------------------------------------------------------------------
extern "C" void kernel_launch(void* const* d_in, const int* in_sizes, int n_in,
                              void* d_out, int out_size, void* d_ws, size_t ws_size,
                              hipStream_t stream) {
    (void)in_sizes; (void)n_in; (void)out_size; (void)ws_size;

    const float* input_q   = (const float*)d_in[0];
    const float* input_k   = (const float*)d_in[1];
    const float* input_v   = (const float*)d_in[2];
    const float* Wq        = (const float*)d_in[3];
    const float* bq        = (const float*)d_in[4];
    const float* Wk        = (const float*)d_in[5];
    const float* bk        = (const float*)d_in[6];
    const float* Wv        = (const float*)d_in[7];
    const float* bv        = (const float*)d_in[8];
    const float* emb_table = (const float*)d_in[9];
    const float* att       = (const float*)d_in[10];
    const int*   idx       = (const int*)d_in[11];
    const unsigned char* masks = (const unsigned char*)d_in[12];

    float* hidden = (float*)d_out;                               // (B,N,D)
    float* attn   = hidden + (size_t)BB * NN * DD;               // (B,H,N,M)

    const size_t head_elems = (size_t)BB * HH * NN * CC;         // 2M floats each
    float* q_ws = (float*)d_ws;
    float* k_ws = q_ws + head_elems;
    float* v_ws = k_ws + head_elems;

    dim3 gproj(BB * NN / 16, DD / 64);   // (512, 4)
    dim3 blk(128);
    qkv_proj_kernel<<<gproj, blk, 0, stream>>>(input_q, Wq, bq, q_ws);
    qkv_proj_kernel<<<gproj, blk, 0, stream>>>(input_k, Wk, bk, k_ws);
    qkv_proj_kernel<<<gproj, blk, 0, stream>>>(input_v, Wv, bv, v_ws);

    dim3 gattn(NN / 16, HH, BB);         // (64, 4, 8) = 2048 workgroups
    lrpe_attn_kernel<<<gattn, blk, 0, stream>>>(q_ws, k_ws, v_ws, emb_table,
                                                att, idx, masks, hidden, attn);
}